// VQEmbeddingEMA_1984274891175
// MI455X (gfx1250) — compile-verified
//
#include <hip/hip_runtime.h>
#include <hip/hip_bf16.h>

// Problem constants
#define BB 16
#define CC 256
#define HH 32
#define WW 32
#define NN 4
#define MM 1024
#define DD 64
#define PP (BB * HH * WW)              // 16384 points per group
#define OUT_ELEMS (BB * CC * HH * WW)  // 4194304
#define PAD_ROW 68                     // 64 floats + 4 pad -> LDS row stride = 4 banks

typedef __attribute__((ext_vector_type(2))) float v2f;
typedef __attribute__((ext_vector_type(8))) float v8f;
typedef __attribute__((ext_vector_type(4))) unsigned int u32x4;
typedef __attribute__((ext_vector_type(8))) unsigned int u32x8;

__device__ __forceinline__ v8f wmma_f32_4(v2f a, v2f b, v8f c) {
  return __builtin_amdgcn_wmma_f32_16x16x4_f32(false, a, false, b, (short)0, c,
                                               false, false);
}

// ---------------------------------------------------------------------------
// Kernel 1: precompute ||e||^2 per code, zero counts + loss accumulator
// ---------------------------------------------------------------------------
__global__ __launch_bounds__(256) void vq_prep_kernel(const float* __restrict__ emb,
                                                      float* __restrict__ e2,
                                                      float* __restrict__ counts,
                                                      float* __restrict__ loss_acc) {
  int i = blockIdx.x * 256 + threadIdx.x;  // 0..4095 over (n,m)
  if (i < NN * MM) {
    const float* row = emb + (size_t)i * DD;
    float s = 0.0f;
    #pragma unroll 8
    for (int d = 0; d < DD; ++d) s += row[d] * row[d];
    e2[i] = s;
    counts[i] = 0.0f;
  }
  if (i == 0) *loss_acc = 0.0f;
}

// ---------------------------------------------------------------------------
// Kernel 2: fused distance + argmin via V_WMMA_F32_16X16X4_F32
//   grid = (128, 4): blockIdx.y = n, blockIdx.x covers 128 P-rows (8 waves x 16)
//   Dynamic LDS: full codebook for group n, padded rows (1024 * 68 floats),
//   staged by the Tensor Data Mover with hardware row padding (TENSORcnt).
// ---------------------------------------------------------------------------
__global__ __launch_bounds__(256) void vq_argmin_kernel(const float* __restrict__ x,
                                                        const float* __restrict__ emb,
                                                        const float* __restrict__ e2,
                                                        int* __restrict__ idx_out,
                                                        float* __restrict__ idx_f_out,
                                                        float* __restrict__ counts) {
  extern __shared__ float smem[];  // MM * PAD_ROW floats (272 KB)
  const int n = blockIdx.y;
  const int tid = threadIdx.x;

  // ---- Stage embedding[n] into LDS via TDM: one 2D descriptor, row padding
  //      64 -> 68 floats done in hardware (pad_interval=64 DW, pad_amount=4 DW).
  const float* En = emb + (size_t)n * MM * DD;
  if (tid < 32) {  // wave 0 only (execz branch skips other waves)
    unsigned long long ga = (unsigned long long)(const void*)En;
    u32x4 g0;
    g0[0] = 1u;                                      // count=1 (valid), no gather
    g0[1] = 0u;                                      // lds_addr = 0 (dyn-LDS base)
    g0[2] = (unsigned int)ga;                        // global_addr[31:0]
    g0[3] = (unsigned int)(ga >> 32) | (2u << 30);   // global_addr[56:32] | type=2
    u32x8 g1;
    g1[0] = (2u << 16)       // data_size = 4 bytes
          | (1u << 20)       // pad_enable
          | (5u << 22)       // pad_interval: 64 DWORDs
          | (3u << 25);      // pad_amount:  4 DWORDs
    g1[1] = (unsigned int)DD << 16;   // tensor_dim0 = 64 elems (bits [79:48] lo)
    g1[2] = (unsigned int)MM << 16;   // tensor_dim1 = 1024    (bits [111:80] lo)
    g1[3] = (unsigned int)DD << 16;   // tile_dim0 = 64        (bits [127:112])
    g1[4] = (unsigned int)MM;         // tile_dim1 = 1024, tile_dim2 = 0
    g1[5] = (unsigned int)DD;         // tensor_dim0_stride = 64 elems
    g1[6] = 0u;                       // stride hi / tensor_dim1_stride lo
    g1[7] = 0u;
    asm volatile("tensor_load_to_lds %0, %1" :: "s"(g0), "s"(g1) : "memory");
    __builtin_amdgcn_s_wait_tensorcnt(0);
  }
  __syncthreads();

  // ---- Wave tile: 16 P-rows x all 1024 codes ----
  const int wave = tid >> 5;
  const int lane = tid & 31;
  const int j    = lane & 15;   // column slot within tile
  const int h2   = lane >> 4;   // half-wave select (K / row-half split)
  const int p0   = (blockIdx.x * 8 + wave) * 16;
  const int p    = p0 + j;
  // x[b][n*64+d][h][w]; p = b*1024 + h*32 + w
  const int xbase = (p >> 10) * (CC * HH * WW) + (p & 1023) + n * DD * (HH * WW);

  // A operand: 16x4 f32 tiles for 16 K-steps. Lane j holds row p0+j;
  // VGPR pair = K {k0+2h, k0+2h+1} per ISA striping.
  v2f a[16];
  #pragma unroll
  for (int kt = 0; kt < 16; ++kt) {
    int d0 = kt * 4 + 2 * h2;
    v2f av;
    av.x = x[xbase + d0 * (HH * WW)];
    av.y = x[xbase + (d0 + 1) * (HH * WW)];
    a[kt] = av;
  }

  float vmin[8];
  int   vidx[8];
  #pragma unroll
  for (int r = 0; r < 8; ++r) { vmin[r] = 3.4e38f; vidx[r] = 0; }

  const float* e2n = e2 + n * MM;

  // 2 M-tiles per iteration x 2-way K-split accumulators.
  for (int mt = 0; mt < MM / 16; mt += 2) {
    const int mrow0 = mt * 16 + j;
    const int mrow1 = mrow0 + 16;
    const float* brow0 = &smem[mrow0 * PAD_ROW + 2 * h2];
    const float* brow1 = &smem[mrow1 * PAD_ROW + 2 * h2];
    float e2v0 = e2n[mrow0];
    float e2v1 = e2n[mrow1];
    v8f acc0e = {}, acc0o = {}, acc1e = {}, acc1o = {};
    #pragma unroll
    for (int kt = 0; kt < 16; kt += 2) {
      v2f b0e = *(const v2f*)(brow0 + kt * 4);
      v2f b0o = *(const v2f*)(brow0 + kt * 4 + 4);
      v2f b1e = *(const v2f*)(brow1 + kt * 4);
      v2f b1o = *(const v2f*)(brow1 + kt * 4 + 4);
      acc0e = wmma_f32_4(a[kt],     b0e, acc0e);
      acc1e = wmma_f32_4(a[kt],     b1e, acc1e);
      acc0o = wmma_f32_4(a[kt + 1], b0o, acc0o);
      acc1o = wmma_f32_4(a[kt + 1], b1o, acc1o);
    }
    // distance (up to a per-row constant): ||e||^2 - 2 * <x, e>
    #pragma unroll
    for (int r = 0; r < 8; ++r) {
      float dot0 = acc0e[r] + acc0o[r];
      float dot1 = acc1e[r] + acc1o[r];
      float dist0 = fmaf(-2.0f, dot0, e2v0);
      float dist1 = fmaf(-2.0f, dot1, e2v1);
      if (dist0 < vmin[r]) { vmin[r] = dist0; vidx[r] = mrow0; }
      if (dist1 < vmin[r]) { vmin[r] = dist1; vidx[r] = mrow1; }
    }
  }

  // ---- Cross-lane argmin within each 16-lane half (columns), ties -> lower idx ----
  #pragma unroll
  for (int r = 0; r < 8; ++r) {
    float mv = vmin[r];
    int   mi = vidx[r];
    #pragma unroll
    for (int off = 8; off >= 1; off >>= 1) {
      float ov = __shfl_xor(mv, off, 32);
      int   oi = __shfl_xor(mi, off, 32);
      if (ov < mv || (ov == mv && oi < mi)) { mv = ov; mi = oi; }
    }
    if (j == 0) {
      int prow = p0 + r + 8 * h2;   // D-tile row layout: VGPR r holds M=r / M=8+r
      idx_out[n * PP + prow]   = mi;
      idx_f_out[n * PP + prow] = (float)mi;
      atomicAdd(&counts[n * MM + mi], 1.0f);
    }
  }
}

// ---------------------------------------------------------------------------
// Kernel 3: gather quantized vectors -> out, accumulate commitment-loss sum
// ---------------------------------------------------------------------------
__global__ __launch_bounds__(256) void vq_out_loss_kernel(const float* __restrict__ x,
                                                          const float* __restrict__ emb,
                                                          const int* __restrict__ idx,
                                                          float* __restrict__ out,
                                                          float* __restrict__ loss_acc) {
  __shared__ float red[256];
  int i = blockIdx.x * 256 + threadIdx.x;  // flat [B,C,H,W]
  int hw = i & (HH * WW - 1);
  int t  = i >> 10;
  int c  = t & (CC - 1);
  int b  = t >> 8;
  int n  = c >> 6;
  int d  = c & (DD - 1);
  int p  = b * (HH * WW) + hw;
  int code = idx[n * PP + p];
  float q  = emb[((size_t)(n * MM + code)) * DD + d];
  float xv = x[i];
  out[i] = q;  // straight-through forward value == quantized
  float diff = xv - q;
  red[threadIdx.x] = diff * diff;
  __syncthreads();
  for (int st = 128; st > 0; st >>= 1) {
    if (threadIdx.x < st) red[threadIdx.x] += red[threadIdx.x + st];
    __syncthreads();
  }
  if (threadIdx.x == 0) atomicAdd(loss_acc, red[0]);
}

// ---------------------------------------------------------------------------
// Kernel 4: perplexity from usage histogram + final loss scaling
// ---------------------------------------------------------------------------
__global__ __launch_bounds__(128) void vq_final_kernel(const float* __restrict__ counts,
                                                       const float* __restrict__ loss_acc,
                                                       float* __restrict__ loss_out,
                                                       float* __restrict__ perp_out) {
  __shared__ float pshare[4];
  int wave = threadIdx.x >> 5;   // one wave per n
  int lane = threadIdx.x & 31;
  float s = 0.0f;
  for (int m = lane; m < MM; m += 32) {
    float avg = counts[wave * MM + m] * (1.0f / (float)PP);
    s += avg * logf(avg + 1e-10f);
  }
  #pragma unroll
  for (int off = 16; off >= 1; off >>= 1) s += __shfl_xor(s, off, 32);
  if (lane == 0) pshare[wave] = expf(-s);
  __syncthreads();
  if (threadIdx.x == 0) {
    *perp_out = pshare[0] + pshare[1] + pshare[2] + pshare[3];
    *loss_out = 0.25f * (*loss_acc) / (float)OUT_ELEMS;
  }
}

// ---------------------------------------------------------------------------
extern "C" void kernel_launch(void* const* d_in, const int* in_sizes, int n_in,
                              void* d_out, int out_size, void* d_ws, size_t ws_size,
                              hipStream_t stream) {
  const float* x   = (const float*)d_in[0];   // [16,256,32,32]
  const float* emb = (const float*)d_in[1];   // [4,1024,64]

  float* out_f   = (float*)d_out;                    // [4194304] quantized output
  float* loss_o  = out_f + OUT_ELEMS;                // [1]
  float* perp_o  = out_f + OUT_ELEMS + 1;            // [1]
  float* idx_f_o = out_f + OUT_ELEMS + 2;            // [4*16384] indices as float

  // Workspace layout (floats): [0] loss accum, [16..) e2 (4096), then counts
  // (4096), then int indices (65536).
  float* wsf      = (float*)d_ws;
  float* loss_acc = wsf;
  float* e2       = wsf + 16;
  float* counts   = wsf + 16 + NN * MM;
  int*   idx_i    = (int*)(wsf + 16 + 2 * NN * MM);

  vq_prep_kernel<<<16, 256, 0, stream>>>(emb, e2, counts, loss_acc);

  dim3 grid2(PP / (16 * 8), NN);                        // (128, 4)
  size_t shmem = (size_t)MM * PAD_ROW * sizeof(float);  // 278528 bytes
  vq_argmin_kernel<<<grid2, 256, shmem, stream>>>(x, emb, e2, idx_i, idx_f_o, counts);

  vq_out_loss_kernel<<<OUT_ELEMS / 256, 256, 0, stream>>>(x, emb, idx_i, out_f, loss_acc);

  vq_final_kernel<<<1, 128, 0, stream>>>(counts, loss_acc, loss_o, perp_o);
}